// GraphConvolution_79800492360332
// MI455X (gfx1250) — compile-verified
//
#include <hip/hip_runtime.h>

#define N_NODES 100000
#define NNZV    3200000
#define DIN     256
#define DOUT    256
#define LDS_STRIDE 260   // 260 % 64 == 4 -> conflict-free ds_load_b64 fragment reads

typedef __attribute__((ext_vector_type(2))) float v2f;
typedef __attribute__((ext_vector_type(4))) float v4f;
typedef __attribute__((ext_vector_type(8))) float v8f;

// Native CDNA5 no-return FP32 atomic add at device scope (RMW performed in L2,
// coherent across all WGPs). No VGPR writeback; tracked by STOREcnt and
// drained by the implicit wait-idle of S_ENDPGM.
__device__ __forceinline__ void atomic_add_f32_dev(float* p, float v) {
    asm volatile("global_atomic_add_f32 %0, %1, off scope:SCOPE_DEV"
                 :
                 : "v"(p), "v"(v)
                 : "memory");
}

// ---------------------------------------------------------------------------
// Kernel 1: Y = X @ W   (fp32 WMMA 16x16x4, exact fp32)
// One workgroup (256 thr = 8 waves) per 16-row panel of X. A-panel staged in
// LDS (coalesced b128 loads), each wave computes two 16x16 N-tiles with K as
// the outer loop, accumulators resident in VGPRs (2 x v8f per wave).
// ---------------------------------------------------------------------------
__global__ __launch_bounds__(256) void gemm_xw(const float* __restrict__ X,
                                               const float* __restrict__ W,
                                               float* __restrict__ Y) {
    __shared__ __align__(16) float As[16 * LDS_STRIDE];
    const int m0 = blockIdx.x * 16;
    const int t  = threadIdx.x;

    // Stage A panel (16 x 256 f32 = 16 KB), 4 float4 per thread, coalesced.
#pragma unroll
    for (int i = 0; i < 4; ++i) {
        int idx = t + i * 256;          // float4 slot 0..1023
        int row = idx >> 6;             // 64 float4 per row
        int c4  = (idx & 63) << 2;
        v4f v = *(const v4f*)(X + (m0 + row) * DIN + c4);
        *(v4f*)&As[row * LDS_STRIDE + c4] = v;   // (row*260 + c4) % 4 == 0
    }
    __syncthreads();

    const int wave = t >> 5;
    const int lane = t & 31;
    const int m  = lane & 15;            // A: row within panel, C/D: column
    const int kb = (lane >> 4) << 1;     // K sub-offset for upper half-wave
    const int n0 = wave * 32;            // this wave's two N-tiles

    const float* wp0 = W + n0 + m;       // B fragment col (n == lane&15 == m)
    const float* wp1 = W + n0 + 16 + m;

    v8f acc0 = {};
    v8f acc1 = {};

#pragma unroll 4
    for (int k0 = 0; k0 < DIN; k0 += 4) {
        const int k = k0 + kb;
        // A fragment: lane holds A[m][k], A[m][k+1]  (conflict-free LDS b64)
        v2f a = *(const v2f*)&As[m * LDS_STRIDE + k];
        // B fragments: lane holds B[k][n], B[k+1][n]  (rows of W, 1 KB apart)
        v2f b0, b1;
        b0.x = wp0[k * DOUT];
        b0.y = wp0[(k + 1) * DOUT];
        b1.x = wp1[k * DOUT];
        b1.y = wp1[(k + 1) * DOUT];
        acc0 = __builtin_amdgcn_wmma_f32_16x16x4_f32(false, a, false, b0,
                                                     (short)0, acc0, false, false);
        acc1 = __builtin_amdgcn_wmma_f32_16x16x4_f32(false, a, false, b1,
                                                     (short)0, acc1, false, false);
    }

    // C/D layout: VGPR j -> M = j (lanes 0-15) / j+8 (lanes 16-31), N = lane&15
    const int rbase = m0 + ((lane >> 4) << 3);
#pragma unroll
    for (int j = 0; j < 8; ++j) {
        Y[(rbase + j) * DOUT + n0 + m]      = acc0[j];
        Y[(rbase + j) * DOUT + n0 + 16 + m] = acc1[j];
    }
}

// ---------------------------------------------------------------------------
// Kernel 2: out[r, :] = bias[:]   (float4 broadcast; rerun every call since
// the harness poisons d_out once and replays the graph)
// ---------------------------------------------------------------------------
__global__ __launch_bounds__(256) void bias_init(const float* __restrict__ bias,
                                                 float* __restrict__ out) {
    int tid = blockIdx.x * blockDim.x + threadIdx.x;   // one float4 each
    int row = tid >> 6;
    int c4  = (tid & 63) << 2;
    v4f b = *(const v4f*)(bias + c4);
    *(v4f*)(out + (long long)row * DOUT + c4) = b;
}

// ---------------------------------------------------------------------------
// Kernel 3: out[row, :] += val * Y[col, :]  for every COO edge of both
// Laplacians. 64 threads per edge, float4 gather (L2-hot Y) + 4 native
// device-scope global_atomic_add_f32 into the L2-resident output.
// ---------------------------------------------------------------------------
__global__ __launch_bounds__(256) void edge_scatter(
    const int*   __restrict__ rA, const int*   __restrict__ cA,
    const float* __restrict__ vA,
    const int*   __restrict__ rB, const int*   __restrict__ cB,
    const float* __restrict__ vB,
    const float* __restrict__ Y,  float* __restrict__ out) {

    unsigned int tid = blockIdx.x * blockDim.x + threadIdx.x;
    unsigned int e  = tid >> 6;          // edge index over 2*NNZ
    int g4 = (int)(tid & 63) << 2;       // feature offset (float4 granule)

    int row, col; float val;
    if (e < (unsigned)NNZV) {
        row = rA[e]; col = cA[e]; val = vA[e];
    } else {
        unsigned int e2 = e - (unsigned)NNZV;
        row = rB[e2]; col = cB[e2]; val = vB[e2];
    }

    v4f y = *(const v4f*)(Y + (long long)col * DOUT + g4);
    float* o = out + (long long)row * DOUT + g4;
    atomic_add_f32_dev(o + 0, val * y.x);
    atomic_add_f32_dev(o + 1, val * y.y);
    atomic_add_f32_dev(o + 2, val * y.z);
    atomic_add_f32_dev(o + 3, val * y.w);
}

// ---------------------------------------------------------------------------
extern "C" void kernel_launch(void* const* d_in, const int* in_sizes, int n_in,
                              void* d_out, int out_size, void* d_ws, size_t ws_size,
                              hipStream_t stream) {
    const float* X    = (const float*)d_in[0];
    const int*   Lr   = (const int*)  d_in[1];
    const int*   Lc   = (const int*)  d_in[2];
    const float* Lv   = (const float*)d_in[3];
    const int*   L3r  = (const int*)  d_in[4];
    const int*   L3c  = (const int*)  d_in[5];
    const float* L3v  = (const float*)d_in[6];
    const float* W    = (const float*)d_in[7];
    const float* bias = (const float*)d_in[8];

    float* out = (float*)d_out;
    float* Y   = (float*)d_ws;          // 100000 * 256 * 4 = 102.4 MB scratch

    // 1) Y = X @ W  (WMMA fp32)
    gemm_xw<<<N_NODES / 16, 256, 0, stream>>>(X, W, Y);

    // 2) out = bias (broadcast)
    bias_init<<<(N_NODES * 64) / 256, 256, 0, stream>>>(bias, out);

    // 3) out += L @ Y + L3 @ Y  (edge scatter with f32 atomics)
    unsigned long long total = 2ULL * NNZV * 64ULL;
    edge_scatter<<<(unsigned int)(total / 256), 256, 0, stream>>>(
        Lr, Lc, Lv, L3r, L3c, L3v, Y, out);
}